// GNN_Module_65429531787946
// MI455X (gfx1250) — compile-verified
//
#include <hip/hip_runtime.h>
#include <hip/hip_bf16.h>

// ---------------------------------------------------------------------------
// GNN (DenseSAGE x3 + BN + dilated adjacency) for MI455X / gfx1250, fp32 WMMA.
// B=8, N=2048, D=H=E=128.
//
// Roofline: adjacency (134 MB) dominates traffic; adj_d fits in the 192 MB L2
// so the 3 SpMM passes stream from L2. LDS-staged tiles give 8x A-reuse across
// the 8 waves of a block; fp32 V_WMMA_16X16X4 does all GEMM-shaped math.
// ---------------------------------------------------------------------------

typedef __attribute__((ext_vector_type(2))) float v2f;
typedef __attribute__((ext_vector_type(8))) float v8f;

#define BATCH 8
#define NNODE 2048
#define FDIM  128
#define T_DIL 10
#define K_DIL 2
#define KCH   64            // K-chunk staged in LDS per SpMM iteration

static __device__ __forceinline__ v8f wmma_f32(v2f a, v2f b, v8f c) {
  // D = A(16x4 f32) * B(4x16 f32) + C(16x16 f32)
  return __builtin_amdgcn_wmma_f32_16x16x4_f32(false, a, false, b,
                                               (short)0, c, false, false);
}

// ---------------------------------------------------------------------------
// 1) bulk copy adj -> adj_d (float4 vectorized)
// ---------------------------------------------------------------------------
__global__ __launch_bounds__(256) void copy4_kernel(const float4* __restrict__ src,
                                                    float4* __restrict__ dst,
                                                    size_t n4) {
  size_t i = (size_t)blockIdx.x * blockDim.x + threadIdx.x;
  size_t stride = (size_t)gridDim.x * blockDim.x;
  for (; i < n4; i += stride) dst[i] = src[i];
}

// ---------------------------------------------------------------------------
// 2) sequential dilation: one block (1024 thr = 32 waves) per batch.
//    Row-by-row: block prefix-scan gives neighbor ranks; remove rank%sf==0,
//    zero row entries and symmetric column entries before the next row.
// ---------------------------------------------------------------------------
__global__ __launch_bounds__(1024) void dilate_kernel(float* __restrict__ adjd,
                                                      const int* __restrict__ mask) {
  const int b = blockIdx.x;
  float* A = adjd + (size_t)b * NNODE * NNODE;
  const int* mk = mask + b * NNODE;
  __shared__ int scan[1024];
  const int t = threadIdx.x;

  for (int n = 0; n < NNODE; ++n) {
    const size_t rowoff = (size_t)n * NNODE;
    float r0 = A[rowoff + 2 * t];
    float r1 = A[rowoff + 2 * t + 1];
    int nb0 = (r0 > 0.f) ? 1 : 0;
    int nb1 = (r1 > 0.f) ? 1 : 0;
    scan[t] = nb0 + nb1;
    __syncthreads();
    // Hillis-Steele inclusive scan over 1024 thread-partials
    for (int off = 1; off < 1024; off <<= 1) {
      int add = (t >= off) ? scan[t - off] : 0;
      __syncthreads();
      scan[t] += add;
      __syncthreads();
    }
    const int num = scan[1023];
    const int excl = scan[t] - (nb0 + nb1);
    const int sf = (num > T_DIL) ? (num + K_DIL - 1) / K_DIL : 1;
    const bool valid = (mk[n] != 0) && (num > 1);
    const int rank0 = excl + nb0;             // 1-based inclusive rank
    const int rank1 = excl + nb0 + nb1;
    const bool rm0 = valid && nb0 && ((rank0 % sf) == 0);
    const bool rm1 = valid && nb1 && ((rank1 % sf) == 0);
    if (rm0) { A[rowoff + 2 * t]     = 0.f; A[(size_t)(2 * t)     * NNODE + n] = 0.f; }
    if (rm1) { A[rowoff + 2 * t + 1] = 0.f; A[(size_t)(2 * t + 1) * NNODE + n] = 0.f; }
    __threadfence();
    __syncthreads();
  }
}

// ---------------------------------------------------------------------------
// 3) reciprocal row degree: invdeg = 1 / max(rowsum, 1). Stored reciprocal so
//    the SpMM epilogue is a multiply, not an IEEE divide chain.
// ---------------------------------------------------------------------------
__global__ __launch_bounds__(256) void invdeg_kernel(const float* __restrict__ adjd,
                                                     float* __restrict__ invdeg) {
  const int row = blockIdx.x;                       // 0 .. B*N-1
  const float* r = adjd + (size_t)row * NNODE;
  float s = 0.f;
  for (int j = threadIdx.x; j < NNODE; j += 256) s += r[j];
  __shared__ float red[256];
  red[threadIdx.x] = s;
  __syncthreads();
  for (int off = 128; off > 0; off >>= 1) {
    if (threadIdx.x < off) red[threadIdx.x] += red[threadIdx.x + off];
    __syncthreads();
  }
  if (threadIdx.x == 0) invdeg[row] = 1.0f / fmaxf(red[0], 1.0f);
}

// ---------------------------------------------------------------------------
// 4) SpMM aggregation via fp32 WMMA with LDS-staged tiles.
//    Block = (b, 16-row tile) x all 128 features; 8 waves each own a 16-col
//    slice. Per K-chunk of 64: stage A(16x64) + H(64x128) in LDS (padded
//    strides for conflict-free reads), then 16 WMMA steps per wave.
//    A global traffic drops 8x vs per-wave loads.
// ---------------------------------------------------------------------------
__global__ __launch_bounds__(256) void spmm_kernel(const float* __restrict__ adjd,
                                                   const float* __restrict__ h,
                                                   const float* __restrict__ invdeg,
                                                   float* __restrict__ agg) {
  const int blk = blockIdx.x;          // B*N/16 = 1024 blocks
  const int b  = blk >> 7;
  const int it = blk & 127;
  const int t    = threadIdx.x;
  const int w    = t >> 5;             // wave -> feature tile (0..7)
  const int lane = t & 31;
  const int mrow  = lane & 15;
  const int khalf = (lane >> 4) * 2;   // {0,1} or {2,3} K-pair (A/B layouts)

  const float* Ab = adjd + (size_t)b * NNODE * NNODE + (size_t)(it * 16) * NNODE;
  const float* Hb = h + (size_t)b * NNODE * FDIM;

  __shared__ float As[16][KCH + 4];    // stride 68: row-strided reads hit
  __shared__ float Hs[KCH][FDIM + 4];  // distinct banks (68,132 % 64 != 0)

  // cooperative-load index precompute
  const int arow = t >> 4;             // 0..15
  const int ac4  = (t & 15) * 4;       // 0..60 step 4
  const int hsub = t >> 5;             // 0..7
  const int hc4  = (t & 31) * 4;       // 0..124 step 4

  v8f c = {};
  for (int kc = 0; kc < NNODE; kc += KCH) {
    // stage A chunk: 16 x 64 (one float4 per thread)
    *(float4*)&As[arow][ac4] =
        *(const float4*)&Ab[(size_t)arow * NNODE + kc + ac4];
    // stage H chunk: 64 x 128 (eight float4 per thread)
#pragma unroll
    for (int rr = 0; rr < 8; ++rr) {
      const int hrow = rr * 8 + hsub;
      *(float4*)&Hs[hrow][hc4] =
          *(const float4*)&Hb[(size_t)(kc + hrow) * FDIM + hc4];
    }
    // prefetch next chunk into cache while this one computes
    if (kc + KCH < NNODE) {
      __builtin_prefetch(&Ab[(size_t)arow * NNODE + kc + KCH + ac4], 0, 1);
      __builtin_prefetch(&Hb[(size_t)(kc + KCH + 8 * hsub) * FDIM + hc4], 0, 1);
    }
    __syncthreads();

#pragma unroll
    for (int kk = 0; kk < KCH; kk += 4) {
      v2f a, bf;
      a.x = As[mrow][kk + khalf];
      a.y = As[mrow][kk + khalf + 1];
      bf.x = Hs[kk + khalf][w * 16 + mrow];
      bf.y = Hs[kk + khalf + 1][w * 16 + mrow];
      c = wmma_f32(a, bf, c);
    }
    __syncthreads();
  }

  float* OUT = agg + (size_t)b * NNODE * FDIM + (size_t)(it * 16) * FDIM + w * 16;
  const float* dg = invdeg + b * NNODE + it * 16;
  const int col = lane & 15;
  const int rbase = (lane >> 4) * 8;
#pragma unroll
  for (int r = 0; r < 8; ++r) {
    const int row = rbase + r;
    OUT[(size_t)row * FDIM + col] = c[r] * dg[row];
  }
}

// ---------------------------------------------------------------------------
// 5) fused lin_rel + lin_root + bias + L2-normalize + mask + ReLU
//    Block = 16 rows x 128 cols; wave w owns 16-col slice; K = 128 via WMMA.
// ---------------------------------------------------------------------------
__global__ __launch_bounds__(256) void sage_post_kernel(const float* __restrict__ agg,
                                                        const float* __restrict__ xin,
                                                        const float* __restrict__ Wr,
                                                        const float* __restrict__ Wx,
                                                        const float* __restrict__ bias,
                                                        const int* __restrict__ mask,
                                                        float* __restrict__ outp) {
  const int blk = blockIdx.x;          // B*N/16 = 1024 blocks
  const int b  = blk >> 7;
  const int it = blk & 127;
  const int w    = threadIdx.x >> 5;   // wave -> col tile
  const int lane = threadIdx.x & 31;

  const size_t rowoff = (size_t)b * NNODE * FDIM + (size_t)(it * 16) * FDIM;
  const float* AG = agg + rowoff;
  const float* X  = xin + rowoff;
  const int mrow  = lane & 15;
  const int khalf = (lane >> 4) * 2;

  v8f c = {};
  for (int kk = 0; kk < FDIM; kk += 4) {
    v2f a, bf;
    // agg @ Wr
    a.x = AG[(size_t)mrow * FDIM + kk + khalf];
    a.y = AG[(size_t)mrow * FDIM + kk + khalf + 1];
    bf.x = Wr[(size_t)(kk + khalf) * FDIM + w * 16 + mrow];
    bf.y = Wr[(size_t)(kk + khalf + 1) * FDIM + w * 16 + mrow];
    c = wmma_f32(a, bf, c);
    // x @ Wx (accumulate into same C)
    a.x = X[(size_t)mrow * FDIM + kk + khalf];
    a.y = X[(size_t)mrow * FDIM + kk + khalf + 1];
    bf.x = Wx[(size_t)(kk + khalf) * FDIM + w * 16 + mrow];
    bf.y = Wx[(size_t)(kk + khalf + 1) * FDIM + w * 16 + mrow];
    c = wmma_f32(a, bf, c);
  }

  __shared__ float tile[16][FDIM + 1];
  __shared__ float rnorm[16];
  const int col = w * 16 + (lane & 15);
  const int rbase = (lane >> 4) * 8;
#pragma unroll
  for (int r = 0; r < 8; ++r) tile[rbase + r][col] = c[r] + bias[col];
  __syncthreads();

  if (threadIdx.x < 16) {
    float s = 0.f;
    for (int d = 0; d < FDIM; ++d) {
      float v = tile[threadIdx.x][d];
      s += v * v;
    }
    rnorm[threadIdx.x] = 1.0f / fmaxf(sqrtf(s), 1e-12f);  // reciprocal once
  }
  __syncthreads();

  float* O = outp + rowoff;
  const int* mk = mask + b * NNODE + it * 16;
  for (int idx = threadIdx.x; idx < 16 * FDIM; idx += 256) {
    const int r = idx >> 7;
    const int d = idx & 127;
    float v = tile[r][d] * rnorm[r] * (float)mk[r];
    O[(size_t)r * FDIM + d] = fmaxf(v, 0.f);   // ReLU after mask
  }
}

// ---------------------------------------------------------------------------
// 6) BatchNorm (training mode, biased var) over B*N rows
// ---------------------------------------------------------------------------
__global__ __launch_bounds__(256) void bn_zero_kernel(float* __restrict__ stats) {
  if (threadIdx.x < 2 * FDIM) stats[threadIdx.x] = 0.f;
}

__global__ __launch_bounds__(128) void bn_reduce_kernel(const float* __restrict__ pre,
                                                        float* __restrict__ stats) {
  const int d = threadIdx.x;                       // feature
  const int rows_per_blk = (BATCH * NNODE) / 256;  // grid = 256 blocks -> 64 rows
  const int r0 = blockIdx.x * rows_per_blk;
  float s1 = 0.f, s2 = 0.f;
  for (int r = 0; r < rows_per_blk; ++r) {
    float v = pre[(size_t)(r0 + r) * FDIM + d];
    s1 += v;
    s2 += v * v;
  }
  atomicAdd(&stats[d], s1);
  atomicAdd(&stats[FDIM + d], s2);
}

__global__ __launch_bounds__(256) void bn_apply_kernel(const float* __restrict__ pre,
                                                       const float* __restrict__ stats,
                                                       const float* __restrict__ g,
                                                       const float* __restrict__ be,
                                                       float* __restrict__ h) {
  const size_t total = (size_t)BATCH * NNODE * FDIM;
  size_t i = (size_t)blockIdx.x * blockDim.x + threadIdx.x;
  if (i >= total) return;
  const int d = (int)(i & (FDIM - 1));
  const float inv_m = 1.0f / (float)(BATCH * NNODE);
  const float m = stats[d] * inv_m;
  const float var = stats[FDIM + d] * inv_m - m * m;
  h[i] = (pre[i] - m) * rsqrtf(var + 1e-5f) * g[d] + be[d];
}

// ---------------------------------------------------------------------------
// 7) final linear: out = (concat(h1,h2,h3) * mf) @ Wl + bl, masked. K = 384.
// ---------------------------------------------------------------------------
__global__ __launch_bounds__(256) void final_kernel(const float* __restrict__ h1,
                                                    const float* __restrict__ h2,
                                                    const float* __restrict__ h3,
                                                    const float* __restrict__ Wl,
                                                    const float* __restrict__ bl,
                                                    const int* __restrict__ mask,
                                                    float* __restrict__ out) {
  const int blk = blockIdx.x;          // 1024 blocks, 16 rows each
  const int b  = blk >> 7;
  const int it = blk & 127;
  const int w    = threadIdx.x >> 5;
  const int lane = threadIdx.x & 31;
  const int mrow  = lane & 15;
  const int khalf = (lane >> 4) * 2;

  const size_t rowoff = (size_t)b * NNODE * FDIM + (size_t)(it * 16) * FDIM;
  const int* mk = mask + b * NNODE + it * 16;
  const float mf_r = (float)mk[mrow];

  v8f c = {};
  for (int kk = 0; kk < 3 * FDIM; kk += 4) {
    const float* H = (kk < FDIM) ? h1 : (kk < 2 * FDIM) ? h2 : h3;
    const int kl = kk & (FDIM - 1);
    v2f a, bf;
    a.x = H[rowoff + (size_t)mrow * FDIM + kl + khalf] * mf_r;
    a.y = H[rowoff + (size_t)mrow * FDIM + kl + khalf + 1] * mf_r;
    bf.x = Wl[(size_t)(kk + khalf) * FDIM + w * 16 + mrow];
    bf.y = Wl[(size_t)(kk + khalf + 1) * FDIM + w * 16 + mrow];
    c = wmma_f32(a, bf, c);
  }

  const int col = w * 16 + (lane & 15);
  const int rbase = (lane >> 4) * 8;
#pragma unroll
  for (int r = 0; r < 8; ++r) {
    const int row = rbase + r;
    out[rowoff + (size_t)row * FDIM + col] = (c[r] + bl[col]) * (float)mk[row];
  }
}

// ---------------------------------------------------------------------------
// host launcher
// ---------------------------------------------------------------------------
extern "C" void kernel_launch(void* const* d_in, const int* in_sizes, int n_in,
                              void* d_out, int out_size, void* d_ws, size_t ws_size,
                              hipStream_t stream) {
  const float* x    = (const float*)d_in[0];
  const float* adj  = (const float*)d_in[1];
  const int*   mask = (const int*)d_in[2];
  const float* Wr[3] = {(const float*)d_in[3],  (const float*)d_in[8],  (const float*)d_in[13]};
  const float* Wx[3] = {(const float*)d_in[4],  (const float*)d_in[9],  (const float*)d_in[14]};
  const float* bv[3] = {(const float*)d_in[5],  (const float*)d_in[10], (const float*)d_in[15]};
  const float* gv[3] = {(const float*)d_in[6],  (const float*)d_in[11], (const float*)d_in[16]};
  const float* bev[3]= {(const float*)d_in[7],  (const float*)d_in[12], (const float*)d_in[17]};
  const float* Wl   = (const float*)d_in[18];
  const float* bl   = (const float*)d_in[19];

  const size_t adj_elems = (size_t)BATCH * NNODE * NNODE;   // 33.5M floats
  const size_t feat_elems = (size_t)BATCH * NNODE * FDIM;   // 2.1M floats

  float* ws    = (float*)d_ws;
  float* adjd  = ws;
  float* ideg  = adjd + adj_elems;
  float* agg   = ideg + (size_t)BATCH * NNODE;
  float* pre   = agg + feat_elems;
  float* h[3];
  h[0] = pre + feat_elems;
  h[1] = h[0] + feat_elems;
  h[2] = h[1] + feat_elems;
  float* stats = h[2] + feat_elems;

  // adj -> adj_d, then sequential dilation + reciprocal degrees
  copy4_kernel<<<8192, 256, 0, stream>>>((const float4*)adj, (float4*)adjd, adj_elems / 4);
  dilate_kernel<<<BATCH, 1024, 0, stream>>>(adjd, mask);
  invdeg_kernel<<<BATCH * NNODE, 256, 0, stream>>>(adjd, ideg);

  const float* hin = x;
  for (int L = 0; L < 3; ++L) {
    spmm_kernel<<<1024, 256, 0, stream>>>(adjd, hin, ideg, agg);
    sage_post_kernel<<<1024, 256, 0, stream>>>(agg, hin, Wr[L], Wx[L], bv[L], mask, pre);
    bn_zero_kernel<<<1, 256, 0, stream>>>(stats);
    bn_reduce_kernel<<<256, 128, 0, stream>>>(pre, stats);
    bn_apply_kernel<<<(unsigned)((feat_elems + 255) / 256), 256, 0, stream>>>(pre, stats, gv[L], bev[L], h[L]);
    hin = h[L];
  }

  final_kernel<<<1024, 256, 0, stream>>>(h[0], h[1], h[2], Wl, bl, mask, (float*)d_out);
}